// Model_74388833566845
// MI455X (gfx1250) — compile-verified
//
#include <hip/hip_runtime.h>

typedef __attribute__((ext_vector_type(2))) float v2f;
typedef __attribute__((ext_vector_type(8))) float v8f;

#define B_    512
#define T_    512
#define H_    100
#define DI_   101
#define DO_   2
#define HP    112          // H padded to 7 tiles of 16
#define HS    116          // h_lds row stride (bank-conflict-free, 8B aligned)
#define KIP   104          // DI padded to 26 k-steps of 4
#define IS    108          // in_lds row stride
#define MB    16           // batch rows per workgroup
#define NWAVE 7
#define NTHREADS (NWAVE * 32)
#define DT_TAU 0.1f

__device__ __forceinline__ float retanh(float x) { return fmaxf(tanhf(x), 0.0f); }

__global__ __launch_bounds__(NTHREADS) void ctrnn_fused_scan(
    const float* __restrict__ input,   // [B,T,DI]
    const float* __restrict__ noise,   // [B,T,H]
    const float* __restrict__ Wx,      // [H,DI]
    const float* __restrict__ bah,     // [H]
    const float* __restrict__ Wh,      // [H,H]
    const float* __restrict__ Why,     // [DO,H]
    const float* __restrict__ ah0,     // [H]
    float* __restrict__ out,           // [B,T,DO]
    float* __restrict__ hstore)        // [B,T,H]
{
    __shared__ float h_lds[2][MB][HS];   // hidden state tile, double buffered
    __shared__ float in_lds[MB][IS];     // input tile staged per step
    __shared__ float why_lds[DO_][HP];   // output weights

    const int tid  = threadIdx.x;
    const int lane = tid & 31;
    const int wave = tid >> 5;          // 0..6 -> N-tile
    const int col  = lane & 15;         // M for A-frags, N within tile for B/C
    const int hi   = lane >> 4;         // half-wave (K sub-offset)
    const int n    = wave * 16 + col;   // global column 0..111
    const bool nvalid = (n < H_);
    const int b0   = blockIdx.x * MB;

    // ---- preload loop-invariant B-fragments (weights) into registers ----
    // recurrent: B[k][n] = Wh[n][k]   (h @ Wh^T)
    float brx[25], bry[25];
    for (int kk = 0; kk < 25; ++kk) {
        int k0 = 4 * kk + 2 * hi;
        brx[kk] = nvalid ? Wh[n * H_ + k0]     : 0.0f;
        bry[kk] = nvalid ? Wh[n * H_ + k0 + 1] : 0.0f;
    }
    // input proj: B[i][n] = Wx[n][i]  (input @ Wx^T), DI=101 padded to 104
    float bxx[26], bxy[26];
    for (int kk = 0; kk < 26; ++kk) {
        int k0 = 4 * kk + 2 * hi;
        bxx[kk] = (nvalid && k0     < DI_) ? Wx[n * DI_ + k0]     : 0.0f;
        bxy[kk] = (nvalid && k0 + 1 < DI_) ? Wx[n * DI_ + k0 + 1] : 0.0f;
    }

    const float bias  = nvalid ? bah[n] : 0.0f;
    const float ahini = nvalid ? ah0[n] : 0.0f;
    float ah[8];
    for (int r = 0; r < 8; ++r) ah[r] = ahini;   // rows m = r + 8*hi, column n

    // ---- init h_lds[0] = retanh(ah0) broadcast over rows; load Why ----
    for (int idx = tid; idx < MB * HP; idx += NTHREADS) {
        int r = idx / HP, c = idx % HP;
        h_lds[0][r][c] = (c < H_) ? retanh(ah0[c]) : 0.0f;
    }
    for (int idx = tid; idx < DO_ * HP; idx += NTHREADS) {
        int d = idx / HP, c = idx % HP;
        why_lds[d][c] = (c < H_) ? Why[d * H_ + c] : 0.0f;
    }
    __syncthreads();

    int cur = 0;
    for (int t = 0; t < T_; ++t) {
        // stage input rows for step t (read input exactly once overall)
        for (int idx = tid; idx < MB * KIP; idx += NTHREADS) {
            int r = idx / KIP, i = idx % KIP;
            in_lds[r][i] = (i < DI_)
                ? input[((size_t)(b0 + r) * T_ + t) * DI_ + i] : 0.0f;
        }
        // fused output GEMM for the previous step (h_{t-1} sits in h_lds[cur])
        if (t > 0 && tid < MB * DO_) {
            int r = tid & 15, d = tid >> 4;
            float s = 0.0f;
            for (int c2 = 0; c2 < H_; ++c2) s += why_lds[d][c2] * h_lds[cur][r][c2];
            out[((size_t)(b0 + r) * T_ + (t - 1)) * DO_ + d] = s;
        }
        __syncthreads();

        // C = input_t @ Wx^T + h_{t-1} @ Wh^T   (one 16x16 tile per wave)
        v8f c = {};
        for (int kk = 0; kk < 26; ++kk) {
            v2f a; a.x = in_lds[col][4 * kk + 2 * hi];
                   a.y = in_lds[col][4 * kk + 2 * hi + 1];
            v2f b; b.x = bxx[kk]; b.y = bxy[kk];
            c = __builtin_amdgcn_wmma_f32_16x16x4_f32(false, a, false, b,
                                                      (short)0, c, false, false);
        }
        for (int kk = 0; kk < 25; ++kk) {
            v2f a; a.x = h_lds[cur][col][4 * kk + 2 * hi];
                   a.y = h_lds[cur][col][4 * kk + 2 * hi + 1];
            v2f b; b.x = brx[kk]; b.y = bry[kk];
            c = __builtin_amdgcn_wmma_f32_16x16x4_f32(false, a, false, b,
                                                      (short)0, c, false, false);
        }

        // state update: ah = 0.9*ah + 0.1*(C + bias); h = retanh(ah) + noise
        const int nxt = cur ^ 1;
        for (int r = 0; r < 8; ++r) {
            int m = r + 8 * hi;
            float nz = nvalid ? noise[((size_t)(b0 + m) * T_ + t) * H_ + n] : 0.0f;
            float a2 = 0.9f * ah[r] + DT_TAU * (c[r] + bias);
            ah[r] = a2;
            float h = nvalid ? (retanh(a2) + nz) : 0.0f;
            h_lds[nxt][m][n] = h;
            if (nvalid) hstore[((size_t)(b0 + m) * T_ + t) * H_ + n] = h;
        }
        __syncthreads();
        cur = nxt;
    }

    // output for the last step
    if (tid < MB * DO_) {
        int r = tid & 15, d = tid >> 4;
        float s = 0.0f;
        for (int c2 = 0; c2 < H_; ++c2) s += why_lds[d][c2] * h_lds[cur][r][c2];
        out[((size_t)(b0 + r) * T_ + (T_ - 1)) * DO_ + d] = s;
    }
}

extern "C" void kernel_launch(void* const* d_in, const int* in_sizes, int n_in,
                              void* d_out, int out_size, void* d_ws, size_t ws_size,
                              hipStream_t stream) {
    (void)in_sizes; (void)n_in; (void)out_size; (void)d_ws; (void)ws_size;
    const float* input = (const float*)d_in[0];
    const float* noise = (const float*)d_in[1];
    const float* Wx    = (const float*)d_in[2];
    const float* bah   = (const float*)d_in[3];
    const float* Wh    = (const float*)d_in[4];
    const float* Why   = (const float*)d_in[5];
    const float* ah0   = (const float*)d_in[6];

    float* out    = (float*)d_out;                       // [B,T,DO]
    float* hstore = out + (size_t)B_ * T_ * DO_;         // [B,T,H]

    ctrnn_fused_scan<<<B_ / MB, NTHREADS, 0, stream>>>(
        input, noise, Wx, bah, Wh, Why, ah0, out, hstore);
}